// EdgePredict_89807766159588
// MI455X (gfx1250) — compile-verified
//
#include <hip/hip_runtime.h>
#include <hip/hip_bf16.h>
#include <math.h>

// ---------------------------------------------------------------------------
// EdgePredict: out = softmax( (h[src] ⊙ h[dst]) @ Wc + bc ),  Wc = W1 W2 W3
// (the reference MLP has no nonlinearity, so it collapses to one affine map).
// GEMM runs on V_WMMA_F32_16X16X4_F32 (full fp32), one wave per 16 edges.
// ---------------------------------------------------------------------------

typedef __attribute__((ext_vector_type(2))) float v2f;
typedef __attribute__((ext_vector_type(8))) float v8f;

#define HD 256           // embedding dim
// float offsets inside workspace
#define WS_W23   0       // [128][2]
#define WS_BC2   256     // [2]   b2@W3 + b3
#define WS_BC    260     // [2]   final collapsed bias
#define WS_BFRAG 512     // [64 ksteps][32 lanes][2 elems] packed B fragments
#define WS_FLOATS 4608

// ---- prep1: W23 = W2 @ W3 (128x2), bc2 = b2@W3 + b3 ------------------------
__global__ void prep1_kernel(const float* __restrict__ W2, const float* __restrict__ W3,
                             const float* __restrict__ b2, const float* __restrict__ b3,
                             float* __restrict__ ws) {
    int t = threadIdx.x;            // 256 threads: t = k*2 + n
    int k = t >> 1, n = t & 1;
    float s = 0.f;
    for (int j = 0; j < 64; ++j) s += W2[k * 64 + j] * W3[j * 2 + n];
    ws[WS_W23 + k * 2 + n] = s;
    if (t < 2) {
        float b = 0.f;
        for (int j = 0; j < 64; ++j) b += b2[j] * W3[j * 2 + t];
        ws[WS_BC2 + t] = b + b3[t];
    }
}

// ---- prep2: pack Wc = W1 @ W23 into WMMA B-fragment order; bc --------------
// B layout (f32 4x16): lane L, elem r -> K = r + (L>=16 ? 2 : 0), N = L&15.
// K-permutation: K-step kk, lanes<16 cover cols kk*2+r, lanes>=16 cover 128+kk*2+r.
__global__ void prep2_kernel(const float* __restrict__ W1, const float* __restrict__ b1,
                             float* __restrict__ ws) {
    int t = blockIdx.x * 256 + threadIdx.x;   // 4096 threads
    int kk  = t >> 6;          // 0..63
    int rem = t & 63;
    int L   = rem >> 1;        // lane 0..31
    int r   = rem & 1;         // elem 0..1
    int N   = L & 15;
    int hi  = L >> 4;
    int col = hi * 128 + kk * 2 + r;   // row of Wc (input feature)
    float v = 0.f;
    if (N < 2) {
        for (int k = 0; k < 128; ++k)
            v += W1[col * 128 + k] * ws[WS_W23 + k * 2 + N];
    }
    ws[WS_BFRAG + t] = v;
    if (t < 2) {
        float b = 0.f;
        for (int k = 0; k < 128; ++k) b += b1[k] * ws[WS_W23 + k * 2 + t];
        ws[WS_BC + t] = b + ws[WS_BC2 + t];
    }
}

// ---- main: gather + hadamard + WMMA fp32 GEMM + 2-class softmax ------------
__global__ void __launch_bounds__(256) edge_mlp_kernel(
    const float* __restrict__ h, const long long* __restrict__ edges,
    const float* __restrict__ ws, float* __restrict__ out, int E) {

    __shared__ float sB[4096];            // 16 KB packed B fragments

    int tid = threadIdx.x;
    // cooperative LDS fill (1024 float4 / 256 threads)
    #pragma unroll
    for (int i = 0; i < 4; ++i) {
        int idx = i * 256 + tid;
        ((float4*)sB)[idx] = ((const float4*)(ws + WS_BFRAG))[idx];
    }
    float bc0 = ws[WS_BC + 0];
    float bc1 = ws[WS_BC + 1];
    __syncthreads();

    int wave = tid >> 5;
    int lane = tid & 31;
    int M    = lane & 15;                 // edge row within the 16-edge tile
    int hi   = lane >> 4;                 // 0: cols 0..127, 1: cols 128..255
    int N    = M;                         // output column this lane holds in C/D

    long long tile = (long long)blockIdx.x * 8 + wave;
    long long e0   = tile * 16 + M;
    long long ec   = (e0 < E) ? e0 : (long long)(E - 1);   // clamp, keep EXEC uniform
    long long src  = edges[2 * ec];
    long long dst  = edges[2 * ec + 1];
    const float* hs = h + src * HD + hi * 128;
    const float* hd = h + dst * HD + hi * 128;

    // accumulator preloaded with collapsed bias (same N for all 8 M-rows)
    v8f c;
    float cinit = (N == 0) ? bc0 : ((N == 1) ? bc1 : 0.f);
    #pragma unroll
    for (int r = 0; r < 8; ++r) c[r] = cinit;

    #pragma unroll 4
    for (int kk2 = 0; kk2 < 32; ++kk2) {
        float4 a = *(const float4*)(hs + kk2 * 4);   // global_load_b128
        float4 b = *(const float4*)(hd + kk2 * 4);
        float p0 = a.x * b.x, p1 = a.y * b.y;        // hadamard (fp32 VALU)
        float p2 = a.z * b.z, p3 = a.w * b.w;

        const float* bp = sB + (kk2 * 2) * 64 + lane * 2;   // ds_load_b64
        v2f A0; A0[0] = p0;    A0[1] = p1;
        v2f B0; B0[0] = bp[0]; B0[1] = bp[1];
        v2f A1; A1[0] = p2;     A1[1] = p3;
        v2f B1; B1[0] = bp[64]; B1[1] = bp[65];

        c = __builtin_amdgcn_wmma_f32_16x16x4_f32(false, A0, false, B0,
                                                  (short)0, c, false, false);
        c = __builtin_amdgcn_wmma_f32_16x16x4_f32(false, A1, false, B1,
                                                  (short)0, c, false, false);
    }

    // 2-class softmax: lanes with N=0/1 (lanes 0,1,16,17) hold the logits;
    // partner logit lives one lane over -> xor-1 shuffle.
    #pragma unroll
    for (int r = 0; r < 8; ++r) {
        float ls = c[r];
        float lo = __shfl_xor(ls, 1, 32);
        float p  = 1.0f / (1.0f + __expf(lo - ls));
        long long e = tile * 16 + (r + hi * 8);     // D row: M = r + 8*hi
        if (N < 2 && e < E) out[e * 2 + N] = p;
    }
}

extern "C" void kernel_launch(void* const* d_in, const int* in_sizes, int n_in,
                              void* d_out, int out_size, void* d_ws, size_t ws_size,
                              hipStream_t stream) {
    const float*     h     = (const float*)d_in[0];
    const long long* edges = (const long long*)d_in[1];
    const float*     W1    = (const float*)d_in[2];
    const float*     b1    = (const float*)d_in[3];
    const float*     W2    = (const float*)d_in[4];
    const float*     b2    = (const float*)d_in[5];
    const float*     W3    = (const float*)d_in[6];
    const float*     b3    = (const float*)d_in[7];
    float* out = (float*)d_out;
    float* ws  = (float*)d_ws;

    int E = in_sizes[1] / 2;

    prep1_kernel<<<1, 256, 0, stream>>>(W2, W3, b2, b3, ws);
    prep2_kernel<<<16, 256, 0, stream>>>(W1, b1, ws);

    int nTiles = (E + 15) / 16;
    int blocks = (nTiles + 7) / 8;   // 8 waves (tiles) per 256-thread block
    edge_mlp_kernel<<<blocks, 256, 0, stream>>>(h, edges, ws, out, E);
}